// TableInterpolation_11871289606717
// MI455X (gfx1250) — compile-verified
//
#include <hip/hip_runtime.h>

typedef float f4 __attribute__((ext_vector_type(4)));
typedef float f2 __attribute__((ext_vector_type(2)));

#define GH 4096
#define GW 4096
#define PPT 4  // points per thread

// Unaligned-capable 8-byte pair load: (grid[y][x], grid[y][x+1]).
// gfx1250 VMEM supports unaligned access, so this lowers to one global_load_b64.
static __device__ __forceinline__ f2 ld_pair(const float* __restrict__ p) {
    f2 r;
    __builtin_memcpy((void*)&r, (const void*)p, sizeof(f2));
    return r;
}

__global__ __launch_bounds__(256) void TableInterp_bilerp4(
    const float* __restrict__ xq,     // [N,2] query coords, row-major (row, col)
    const float* __restrict__ grid,   // [H,W] table
    const float* __restrict__ bounds, // [[lo_y, lo_x], [hi_y, hi_x]]
    float* __restrict__ out,          // [N]
    int n)
{
    // Uniform scalars (compiler scalarizes: uniform address, invariant loads).
    const float lo_y = bounds[0];
    const float lo_x = bounds[1];
    const float sy = (float)(GH - 1) / (bounds[2] - bounds[0]);
    const float sx = (float)(GW - 1) / (bounds[3] - bounds[1]);

    const int i0 = (blockIdx.x * blockDim.x + threadIdx.x) * PPT;

    if (i0 + PPT <= n) {
        // Streaming query read: two coalesced b128 loads, non-temporal so the
        // 64MB x-array does not evict the grid from the 192MB L2.
        const f4 p01 = __builtin_nontemporal_load((const f4*)(xq + (size_t)i0 * 2));
        const f4 p23 = __builtin_nontemporal_load((const f4*)(xq + (size_t)i0 * 2 + 4));

        float qy[PPT], qx[PPT];
        qy[0] = sy * (p01.x - lo_y); qx[0] = sx * (p01.y - lo_x);
        qy[1] = sy * (p01.z - lo_y); qx[1] = sx * (p01.w - lo_x);
        qy[2] = sy * (p23.x - lo_y); qx[2] = sx * (p23.y - lo_x);
        qy[3] = sy * (p23.z - lo_y); qx[3] = sx * (p23.w - lo_x);

        f2 top[PPT], bot[PPT];
        float ax[PPT], ay[PPT];
        // Issue all 8 gathers back-to-back (default RT temporal hint: we WANT
        // grid lines to live in L2 — the whole table fits).
        #pragma unroll
        for (int k = 0; k < PPT; ++k) {
            float fy = fminf(fmaxf(floorf(qy[k]), 0.0f), (float)(GH - 2));
            float fx = fminf(fmaxf(floorf(qx[k]), 0.0f), (float)(GW - 2));
            ay[k] = fminf(fmaxf(qy[k] - fy, 0.0f), 1.0f);
            ax[k] = fminf(fmaxf(qx[k] - fx, 0.0f), 1.0f);
            const int idx = (int)fy * GW + (int)fx;
            top[k] = ld_pair(grid + idx);         // (tl, tr)
            bot[k] = ld_pair(grid + idx + GW);    // (bl, br)
        }

        f4 r;
        #pragma unroll
        for (int k = 0; k < PPT; ++k) {
            float t = fmaf(top[k].y - top[k].x, ax[k], top[k].x);
            float b = fmaf(bot[k].y - bot[k].x, ax[k], bot[k].x);
            r[k] = fmaf(b - t, ay[k], t);
        }
        // Streaming result: one b128 non-temporal store (32MB never re-read).
        __builtin_nontemporal_store(r, (f4*)(out + i0));
    } else {
        // Tail (not taken for N = 8388608, kept for generality).
        for (int i = i0; i < n; ++i) {
            f2 p = ld_pair(xq + (size_t)i * 2);
            float qyv = sy * (p.x - lo_y);
            float qxv = sx * (p.y - lo_x);
            float fy = fminf(fmaxf(floorf(qyv), 0.0f), (float)(GH - 2));
            float fx = fminf(fmaxf(floorf(qxv), 0.0f), (float)(GW - 2));
            float ayv = fminf(fmaxf(qyv - fy, 0.0f), 1.0f);
            float axv = fminf(fmaxf(qxv - fx, 0.0f), 1.0f);
            int idx = (int)fy * GW + (int)fx;
            f2 tp = ld_pair(grid + idx);
            f2 bt = ld_pair(grid + idx + GW);
            float t = fmaf(tp.y - tp.x, axv, tp.x);
            float b = fmaf(bt.y - bt.x, axv, bt.x);
            out[i] = fmaf(b - t, ayv, t);
        }
    }
}

extern "C" void kernel_launch(void* const* d_in, const int* in_sizes, int n_in,
                              void* d_out, int out_size, void* d_ws, size_t ws_size,
                              hipStream_t stream) {
    const float* xq     = (const float*)d_in[0];  // [N,2]
    const float* grid   = (const float*)d_in[1];  // [1,H,W,1] -> flat H*W
    const float* bounds = (const float*)d_in[2];  // [2,2]
    float* out = (float*)d_out;

    const int n = in_sizes[0] / 2;                // number of query points
    const int threads = 256;                      // 8 wave32s per workgroup
    const int blocks = (n + threads * PPT - 1) / (threads * PPT);
    hipLaunchKernelGGL(TableInterp_bilerp4, dim3(blocks), dim3(threads), 0, stream,
                       xq, grid, bounds, out, n);
}